// CARAFE_3_pow_norm_206158430925
// MI455X (gfx1250) — compile-verified
//
#include <hip/hip_runtime.h>
#include <hip/hip_bf16.h>

typedef __attribute__((ext_vector_type(2))) float v2f;
typedef __attribute__((ext_vector_type(8))) float v8f;

#define B_   2
#define C_   256
#define HW_  4096   // 64*64
#define CMID 64
#define CENC 100

// ---------------------------------------------------------------------------
// Stage 1: 1x1 conv (256->64) + BN + ReLU as GEMM via V_WMMA_F32_16X16X4_F32
// Per batch: out[64,4096] = compW[64,256] * X[256,4096]
// One wave -> 16(M) x 64(N) tile. Waves: 2 * 4 * 64 = 512.
// ---------------------------------------------------------------------------
__global__ __launch_bounds__(256) void comp_gemm_bn_relu(
    const float* __restrict__ X, const float* __restrict__ Wc,
    const float* __restrict__ gamma, const float* __restrict__ beta,
    const float* __restrict__ mean, const float* __restrict__ var,
    float* __restrict__ W1)
{
    const int lane = threadIdx.x & 31;
    const int wid  = (blockIdx.x * blockDim.x + threadIdx.x) >> 5;
    const int b    = wid >> 8;        // 256 waves per batch
    const int r    = wid & 255;
    const int m0   = (r >> 6) * 16;   // 4 m-tiles
    const int n0   = (r & 63) * 64;   // 64 n-tiles of width 64
    const int half = lane >> 4;
    const int l16  = lane & 15;

    const float* Xb = X + b * (C_ * HW_);
    const float* Arow = Wc + (m0 + l16) * C_;   // A row for this lane (M = m0+l16)

    v8f acc0 = {}, acc1 = {}, acc2 = {}, acc3 = {};

    for (int k = 0; k < C_; k += 4) {
        v2f a;
        a.x = Arow[k + 2 * half];
        a.y = Arow[k + 2 * half + 1];
        const float* B0 = Xb + (k + 2 * half) * HW_;
        const float* B1 = B0 + HW_;

        v2f bf;
        bf.x = B0[n0 + 0 * 16 + l16]; bf.y = B1[n0 + 0 * 16 + l16];
        acc0 = __builtin_amdgcn_wmma_f32_16x16x4_f32(false, a, false, bf, (short)0, acc0, false, false);
        bf.x = B0[n0 + 1 * 16 + l16]; bf.y = B1[n0 + 1 * 16 + l16];
        acc1 = __builtin_amdgcn_wmma_f32_16x16x4_f32(false, a, false, bf, (short)0, acc1, false, false);
        bf.x = B0[n0 + 2 * 16 + l16]; bf.y = B1[n0 + 2 * 16 + l16];
        acc2 = __builtin_amdgcn_wmma_f32_16x16x4_f32(false, a, false, bf, (short)0, acc2, false, false);
        bf.x = B0[n0 + 3 * 16 + l16]; bf.y = B1[n0 + 3 * 16 + l16];
        acc3 = __builtin_amdgcn_wmma_f32_16x16x4_f32(false, a, false, bf, (short)0, acc3, false, false);
    }

    // eval-mode BN + ReLU; C/D layout: VGPR i -> M = i + 8*half, N = l16
    float inv_[8], bias_[8];
    #pragma unroll
    for (int i = 0; i < 8; ++i) {
        const int ch = m0 + i + 8 * half;
        const float iv = gamma[ch] * rsqrtf(var[ch] + 1e-5f);
        inv_[i]  = iv;
        bias_[i] = beta[ch] - mean[ch] * iv;
    }
    float* W1b = W1 + b * (CMID * HW_);
    #pragma unroll
    for (int sub = 0; sub < 4; ++sub) {
        const int s = n0 + sub * 16 + l16;
        const v8f acc = (sub == 0) ? acc0 : (sub == 1) ? acc1 : (sub == 2) ? acc2 : acc3;
        #pragma unroll
        for (int i = 0; i < 8; ++i) {
            const int ch = m0 + i + 8 * half;
            W1b[ch * HW_ + s] = fmaxf(acc[i] * inv_[i] + bias_[i], 0.0f);
        }
    }
}

// ---------------------------------------------------------------------------
// Stage 2: 3x3 conv (64->100, pad 1) + BN as im2col GEMM.
// M=100 (7 tiles of 16, rows >=100 masked), N=4096, K padded to 64*12.
// One wave handles one full image row y (x = 0..63) for one M-tile.
// Per input channel cm: 3 WMMA K-steps with compile-time tap indices.
// Waves: 2 * 7 * 64 = 896.
// ---------------------------------------------------------------------------
__global__ __launch_bounds__(256) void enc_gemm_bn(
    const float* __restrict__ W1, const float* __restrict__ We,
    const float* __restrict__ gamma, const float* __restrict__ beta,
    const float* __restrict__ mean, const float* __restrict__ var,
    float* __restrict__ W2)
{
    const int lane = threadIdx.x & 31;
    const int wid  = (blockIdx.x * blockDim.x + threadIdx.x) >> 5;
    const int b    = wid / 448;
    const int r    = wid % 448;
    const int m0   = (r / 64) * 16;   // 0..96
    const int y    = r % 64;
    const int half = lane >> 4;
    const int l16  = lane & 15;

    const float* W1b = W1 + b * (CMID * HW_);
    const int  ce    = m0 + l16;
    const bool arow_ok = (ce < CENC);
    const float* Arow  = We + min(ce, CENC - 1) * (CMID * 9);

    // BN params for this wave's output channels
    float inv_[8], bias_[8];
    #pragma unroll
    for (int i = 0; i < 8; ++i) {
        const int ch  = m0 + i + 8 * half;
        const int chc = min(ch, CENC - 1);
        const float iv = gamma[chc] * rsqrtf(var[chc] + 1e-5f);
        inv_[i]  = iv;
        bias_[i] = beta[chc] - mean[chc] * iv;
    }

    float* W2b = W2 + b * (CENC * HW_);

    for (int sub = 0; sub < 4; ++sub) {
        const int x = sub * 16 + l16;
        // 9 taps: t = ki*3+kj, source (y+ki-1, x+kj-1); clamped offset + valid
        int  off[9];
        bool val[9];
        #pragma unroll
        for (int ki = 0; ki < 3; ++ki) {
            #pragma unroll
            for (int kj = 0; kj < 3; ++kj) {
                const int t  = ki * 3 + kj;
                const int yy = y + ki - 1, xx = x + kj - 1;
                val[t] = (yy >= 0) && (yy < 64) && (xx >= 0) && (xx < 64);
                off[t] = min(max(yy, 0), 63) * 64 + min(max(xx, 0), 63);
            }
        }
        // per-lane tap selections (compile-time array indices only)
        const int  o0 = half ? off[2] : off[0];  const bool v0 = half ? val[2] : val[0];
        const int  o1 = half ? off[3] : off[1];  const bool v1 = half ? val[3] : val[1];
        const int  o4 = half ? off[6] : off[4];  const bool v4 = half ? val[6] : val[4];
        const int  o5 = half ? off[7] : off[5];  const bool v5 = half ? val[7] : val[5];
        const int  o8 = off[8];                  const bool v8v = val[8] && (half == 0);

        v8f acc = {};
        for (int cm = 0; cm < CMID; ++cm) {
            const float* Bcm = W1b + cm * HW_;
            const float* Acm = Arow + cm * 9;
            // K-step 0: taps {0,1 | 2,3}
            {
                float ax = Acm[2 * half], ay = Acm[2 * half + 1];
                float bx = Bcm[o0],       by = Bcm[o1];
                v2f a; a.x = arow_ok ? ax : 0.0f; a.y = arow_ok ? ay : 0.0f;
                v2f bf; bf.x = v0 ? bx : 0.0f;    bf.y = v1 ? by : 0.0f;
                acc = __builtin_amdgcn_wmma_f32_16x16x4_f32(false, a, false, bf, (short)0, acc, false, false);
            }
            // K-step 1: taps {4,5 | 6,7}
            {
                float ax = Acm[4 + 2 * half], ay = Acm[5 + 2 * half];
                float bx = Bcm[o4],           by = Bcm[o5];
                v2f a; a.x = arow_ok ? ax : 0.0f; a.y = arow_ok ? ay : 0.0f;
                v2f bf; bf.x = v4 ? bx : 0.0f;    bf.y = v5 ? by : 0.0f;
                acc = __builtin_amdgcn_wmma_f32_16x16x4_f32(false, a, false, bf, (short)0, acc, false, false);
            }
            // K-step 2: tap {8 | pad}, rest zero-padded
            {
                float ax = Acm[8];
                float bx = Bcm[o8];
                v2f a; a.x = (arow_ok && half == 0) ? ax : 0.0f; a.y = 0.0f;
                v2f bf; bf.x = v8v ? bx : 0.0f;                  bf.y = 0.0f;
                acc = __builtin_amdgcn_wmma_f32_16x16x4_f32(false, a, false, bf, (short)0, acc, false, false);
            }
        }
        // BN (no relu) + masked store
        const int s = y * 64 + x;
        #pragma unroll
        for (int i = 0; i < 8; ++i) {
            const int ch = m0 + i + 8 * half;
            if (ch < CENC) {
                W2b[ch * HW_ + s] = acc[i] * inv_[i] + bias_[i];
            }
        }
    }
}

// ---------------------------------------------------------------------------
// Stage 3: pixel-shuffle + clamp/pow + softmax(25 taps) + CARAFE reassembly.
// Identity used: nearest x2 upsample + k=5 dil=2 pad=4 unfold  =>
//   out[b,c,2y+ry,2x+rx] = sum_{ki,kj} W[tap] * X[b,c,y+ki-2,x+kj-2] (0 if OOB)
// One thread per output pixel; weights computed once, reused for 256 channels.
// Zero-padding folded into weights (post-softmax), loads stay in-bounds.
// ---------------------------------------------------------------------------
__global__ __launch_bounds__(256) void carafe_apply(
    const float* __restrict__ X, const float* __restrict__ W2,
    const float* __restrict__ power_p, float* __restrict__ out)
{
    const int tid = blockIdx.x * blockDim.x + threadIdx.x;  // 0 .. 2*128*128-1
    const int b   = tid >> 14;
    const int p2  = tid & 16383;
    const int yp  = p2 >> 7, xp = p2 & 127;
    const int y   = yp >> 1, x  = xp >> 1;
    const int subc = (yp & 1) * 2 + (xp & 1);   // pixel-shuffle sub-channel

    // Gather 25 raw kernel values: channel = tap*4 + subc
    const float* W2p = W2 + b * (CENC * HW_) + subc * HW_ + y * 64 + x;
    float w[25];
    #pragma unroll
    for (int t = 0; t < 25; ++t) w[t] = W2p[t * 4 * HW_];

    // clamp -> pow -> softmax over taps
    const float pw = fmaxf(power_p[0], 1e-5f);
    float mx = -3.0e38f;
    #pragma unroll
    for (int t = 0; t < 25; ++t) {
        const float v = __powf(fmaxf(w[t], 1e-5f), pw);
        w[t] = v;
        mx = fmaxf(mx, v);
    }
    float ssum = 0.0f;
    #pragma unroll
    for (int t = 0; t < 25; ++t) { w[t] = __expf(w[t] - mx); ssum += w[t]; }
    const float rs = 1.0f / ssum;

    // clamped source offsets; OOB handled by zeroing effective weight
    int off[25];
    #pragma unroll
    for (int ki = 0; ki < 5; ++ki) {
        #pragma unroll
        for (int kj = 0; kj < 5; ++kj) {
            const int t  = ki * 5 + kj;
            const int yy = y + ki - 2, xx = x + kj - 2;
            const bool v = (yy >= 0) && (yy < 64) && (xx >= 0) && (xx < 64);
            off[t] = min(max(yy, 0), 63) * 64 + min(max(xx, 0), 63);
            w[t]   = v ? w[t] * rs : 0.0f;
        }
    }

    const float* Xb = X + b * (C_ * HW_);
    float* ob = out + b * (C_ * 128 * 128) + yp * 128 + xp;
    for (int c = 0; c < C_; ++c) {
        const float* Xc = Xb + c * HW_;
        float acc = 0.0f;
        #pragma unroll
        for (int t = 0; t < 25; ++t) acc = fmaf(w[t], Xc[off[t]], acc);
        ob[c * (128 * 128)] = acc;
    }
}

// ---------------------------------------------------------------------------
extern "C" void kernel_launch(void* const* d_in, const int* in_sizes, int n_in,
                              void* d_out, int out_size, void* d_ws, size_t ws_size,
                              hipStream_t stream)
{
    const float* X  = (const float*)d_in[0];
    const float* cw = (const float*)d_in[1];
    const float* cg = (const float*)d_in[2];
    const float* cb = (const float*)d_in[3];
    const float* cm = (const float*)d_in[4];
    const float* cv = (const float*)d_in[5];
    const float* ew = (const float*)d_in[6];
    const float* eg = (const float*)d_in[7];
    const float* eb = (const float*)d_in[8];
    const float* em = (const float*)d_in[9];
    const float* ev = (const float*)d_in[10];
    const float* pp = (const float*)d_in[11];
    float* out = (float*)d_out;

    float* w1 = (float*)d_ws;                    // 2*64*4096  floats (2 MB)
    float* w2 = w1 + (size_t)B_ * CMID * HW_;    // 2*100*4096 floats (3.3 MB)

    // Stage 1: 512 waves -> 64 blocks of 256
    comp_gemm_bn_relu<<<64, 256, 0, stream>>>(X, cw, cg, cb, cm, cv, w1);
    // Stage 2: 896 waves -> 112 blocks of 256
    enc_gemm_bn<<<112, 256, 0, stream>>>(w1, ew, eg, eb, em, ev, w2);
    // Stage 3: 32768 output pixels -> 128 blocks of 256
    carafe_apply<<<128, 256, 0, stream>>>(X, w2, pp, out);
}